// _DeformableAttention_31559419691120
// MI455X (gfx1250) — compile-verified
//
#include <hip/hip_runtime.h>
#include <hip/hip_bf16.h>

// ---------------- problem constants ----------------
#define B_  32
#define Q_  300
#define D_  256
#define H_  8
#define DH_ 32
#define L_  4
#define P_  4
#define S_  8500
#define K_  256   // inner dim of every GEMM here

typedef float v2f __attribute__((ext_vector_type(2)));
typedef float v8f __attribute__((ext_vector_type(8)));

// =====================================================================
// Generic fp32 WMMA GEMM:  C[M,N] = A[M,256] * W[256,N] + bias[N]
// Block = 256 threads = 8 waves. Tile: BM=128 (16 rows/wave), BN=64.
// Double-buffered LDS; A staged via GLOBAL_LOAD_ASYNC_TO_LDS_B128
// (ASYNCcnt), B pipelined through registers into a K-pair-interleaved
// LDS layout so every B fragment is a single aligned ds_load_b64.
// Matrix op: V_WMMA_F32_16X16X4_F32.
// =====================================================================
#define BM 128
#define BN 64
#define KT 16
#define NSTEPS (K_ / KT)   // 16

__device__ __forceinline__ void wait_async0() {
#if __has_builtin(__builtin_amdgcn_s_wait_asynccnt)
    __builtin_amdgcn_s_wait_asynccnt(0);
#else
    asm volatile("s_wait_asynccnt 0x0" ::: "memory");
#endif
}

// async copy 16B global -> LDS (per lane), tracked by ASYNCcnt
__device__ __forceinline__ void async_b128(uint32_t ldsoff, const float* g) {
    asm volatile("global_load_async_to_lds_b128 %0, %1, off"
                 :: "v"(ldsoff), "v"(g) : "memory");
}

// A tile stage: 128 rows x 16 cols = 8 KB; 256 thr x 2 x b128
__device__ __forceinline__ void stageA_async(float* Asb, const float* A,
                                             int m0, int k0) {
    const int tid = threadIdx.x;
    const int r   = tid >> 2;           // 0..63
    const int cg  = (tid & 3) * 4;      // 0,4,8,12
    const uint32_t l0 = (uint32_t)(uintptr_t)&Asb[r * KT + cg];
    const uint32_t l1 = (uint32_t)(uintptr_t)&Asb[(r + 64) * KT + cg];
    async_b128(l0, &A[(size_t)(m0 + r) * K_ + k0 + cg]);
    async_b128(l1, &A[(size_t)(m0 + r + 64) * K_ + k0 + cg]);
}

// B tile: 16 rows x 64 cols; registers -> K-pair interleaved LDS
__device__ __forceinline__ float4 loadB(const float* W, int N, int n0, int k0) {
    const int tid = threadIdx.x;
    const int r   = tid >> 4;           // 0..15 (k within tile)
    const int c   = (tid & 15) * 4;     // 0..60
    return *(const float4*)&W[(size_t)(k0 + r) * N + n0 + c];
}
__device__ __forceinline__ void storeB(float* Bsb, float4 v) {
    const int tid = threadIdx.x;
    const int r   = tid >> 4;
    const int c   = (tid & 15) * 4;
    // layout: Bsb[(k>>1)*128 + col*2 + (k&1)]
    float* dst = &Bsb[(r >> 1) * 128 + (r & 1)];
    dst[(c + 0) * 2] = v.x;
    dst[(c + 1) * 2] = v.y;
    dst[(c + 2) * 2] = v.z;
    dst[(c + 3) * 2] = v.w;
}

__device__ __forceinline__ void compute_tile(const float* Asb, const float* Bsb,
                                             v8f acc[4], int wave, int lane) {
    const int half = lane >> 4;               // 0: lanes 0-15, 1: 16-31
    const int mrow = wave * 16 + (lane & 15);
    const int ncol = lane & 15;
#pragma unroll
    for (int kk = 0; kk < KT; kk += 4) {
        const int bk = kk + half * 2;         // even
        const v2f afrag = *(const v2f*)&Asb[mrow * KT + bk];
#pragma unroll
        for (int s = 0; s < 4; ++s) {
            const v2f bfrag =
                *(const v2f*)&Bsb[(bk >> 1) * 128 + (s * 16 + ncol) * 2];
            acc[s] = __builtin_amdgcn_wmma_f32_16x16x4_f32(
                false, afrag, false, bfrag, (short)0, acc[s], false, false);
        }
    }
}

__global__ __launch_bounds__(256)
void gemm_wmma_f32(const float* __restrict__ A,
                   const float* __restrict__ W,
                   const float* __restrict__ bias,
                   float* __restrict__ C,
                   int M, int N)
{
    __shared__ float As[2][BM * KT];   // 2 x 8 KB
    __shared__ float Bs[2][KT * BN];   // 2 x 4 KB

    const int tid  = threadIdx.x;
    const int lane = tid & 31;
    const int wave = tid >> 5;
    const int m0   = blockIdx.x * BM;
    const int n0   = blockIdx.y * BN;

    v8f acc[4];
#pragma unroll
    for (int s = 0; s < 4; ++s) acc[s] = (v8f)0.0f;

    // ---- prologue: stage tile 0
    int buf = 0;
    stageA_async(As[0], A, m0, 0);
    float4 breg = loadB(W, N, n0, 0);
    wait_async0();
    storeB(Bs[0], breg);
    __syncthreads();

    // ---- software-pipelined main loop
    for (int step = 0; step < NSTEPS; ++step) {
        if (step + 1 < NSTEPS) {
            const int kn = (step + 1) * KT;
            stageA_async(As[buf ^ 1], A, m0, kn);   // overlap with compute
            breg = loadB(W, N, n0, kn);
        }
        compute_tile(As[buf], Bs[buf], acc, wave, lane);
        if (step + 1 < NSTEPS) {
            wait_async0();                          // own async A done
            storeB(Bs[buf ^ 1], breg);
            __syncthreads();                        // tile buf^1 visible to all
        }
        buf ^= 1;
    }

    // ---- store D (+bias): VGPR j -> row j (lanes 0-15) / row j+8 (lanes 16-31)
    const int wrow = wave * 16;
    const int rofs = ((lane >> 4) << 3);
    const int ncol = (lane & 15);
#pragma unroll
    for (int s = 0; s < 4; ++s) {
        const int col = n0 + s * 16 + ncol;
        const float bv = bias[col];
#pragma unroll
        for (int j = 0; j < 8; ++j) {
            const int row = m0 + wrow + j + rofs;
            C[(size_t)row * N + col] = acc[s][j] + bv;
        }
    }
}

// =====================================================================
// Sampling kernel: one wave per (b,q,h); lane = channel (DH_=32 = wave32).
// =====================================================================
__global__ __launch_bounds__(256)
void deform_sample(const float* __restrict__ value,   // (B,S,256)
                   const float* __restrict__ off,     // (B,Q,256)
                   const float* __restrict__ aw,      // (B,Q,128)
                   const float* __restrict__ ref,     // (B,Q,4)
                   float* __restrict__ mixed)         // (B,Q,256)
{
    const int lane  = threadIdx.x & 31;
    const int wave  = threadIdx.x >> 5;
    const int gwave = blockIdx.x * 8 + wave;          // 0 .. B*Q*H-1
    const int h  = gwave % H_;
    const int bq = gwave / H_;
    const int q  = bq % Q_;
    const int b  = bq / Q_;

    constexpr int hhA[L_] = {80, 40, 20, 10};
    constexpr int wwA[L_] = {80, 40, 20, 10};
    constexpr int stA[L_] = {0, 6400, 8000, 8400};

    // softmax over 16 attention logits (redundant per lane; cheap)
    const float* awp = aw + (size_t)(b * Q_ + q) * 128 + h * 16;
    float lg[16];
#pragma unroll
    for (int i = 0; i < 16; ++i) lg[i] = awp[i];
    float mx = lg[0];
#pragma unroll
    for (int i = 1; i < 16; ++i) mx = fmaxf(mx, lg[i]);
    float ssum = 0.0f;
#pragma unroll
    for (int i = 0; i < 16; ++i) { lg[i] = __expf(lg[i] - mx); ssum += lg[i]; }
    const float inv = 1.0f / ssum;

    const float* refp = ref + (size_t)(b * Q_ + q) * 4;
    const float rx = refp[0], ry = refp[1];
    const float rw = refp[2] * 0.5f, rh = refp[3] * 0.5f;

    const float* offp  = off + (size_t)(b * Q_ + q) * 256 + h * (L_ * P_ * 2);
    const float* vbase = value + (size_t)b * S_ * D_ + h * DH_ + lane;

    float accv = 0.0f;
#pragma unroll
    for (int l = 0; l < L_; ++l) {
        const int hh = hhA[l], ww = wwA[l], st = stA[l];
#pragma unroll
        for (int p = 0; p < P_; ++p) {
            const float ox = offp[(l * P_ + p) * 2 + 0];
            const float oy = offp[(l * P_ + p) * 2 + 1];
            const float gx = 2.0f * (rx + ox * 0.25f * rw) - 1.0f;
            const float gy = 2.0f * (ry + oy * 0.25f * rh) - 1.0f;
            const float x = ((gx + 1.0f) * (float)ww - 1.0f) * 0.5f;
            const float y = ((gy + 1.0f) * (float)hh - 1.0f) * 0.5f;
            const float x0f = floorf(x), y0f = floorf(y);
            const float lx = x - x0f, ly = y - y0f;
            const int x0 = (int)x0f, y0 = (int)y0f;
            const int x1 = x0 + 1,  y1 = y0 + 1;

            auto sample = [&](int xi, int yi) -> float {
                const bool valid = (xi >= 0) & (xi < ww) & (yi >= 0) & (yi < hh);
                const int xc = min(max(xi, 0), ww - 1);
                const int yc = min(max(yi, 0), hh - 1);
                const float v = vbase[(size_t)(st + yc * ww + xc) * D_];
                return valid ? v : 0.0f;
            };

            const float s00 = sample(x0, y0);
            const float s01 = sample(x1, y0);
            const float s10 = sample(x0, y1);
            const float s11 = sample(x1, y1);
            const float bil = s00 * (1.0f - lx) * (1.0f - ly)
                            + s01 * lx * (1.0f - ly)
                            + s10 * (1.0f - lx) * ly
                            + s11 * lx * ly;
            accv += (lg[l * P_ + p] * inv) * bil;
        }
    }
    mixed[(size_t)(b * Q_ + q) * D_ + h * DH_ + lane] = accv;
}

// =====================================================================
// launch
// =====================================================================
extern "C" void kernel_launch(void* const* d_in, const int* in_sizes, int n_in,
                              void* d_out, int out_size, void* d_ws, size_t ws_size,
                              hipStream_t stream) {
    const float* hidden  = (const float*)d_in[0];   // (B,Q,256)
    const float* encoder = (const float*)d_in[1];   // (B,S,256)
    const float* refpts  = (const float*)d_in[2];   // (B,Q,1,4)
    const float* W_val   = (const float*)d_in[3];
    const float* b_val   = (const float*)d_in[4];
    const float* W_off   = (const float*)d_in[5];
    const float* b_off   = (const float*)d_in[6];
    const float* W_attn  = (const float*)d_in[7];
    const float* b_attn  = (const float*)d_in[8];
    const float* W_out   = (const float*)d_in[9];
    const float* b_out   = (const float*)d_in[10];
    float* out = (float*)d_out;

    // workspace layout (floats)
    float* ws    = (float*)d_ws;
    float* value = ws;                                   // B*S*256  = 69,632,000
    float* offb  = value + (size_t)B_ * S_ * D_;         // B*Q*256
    float* awb   = offb  + (size_t)B_ * Q_ * 256;        // B*Q*128
    float* mixed = awb   + (size_t)B_ * Q_ * 128;        // B*Q*256

    const int MV = B_ * S_;   // 272000 (divisible by 128)
    const int MQ = B_ * Q_;   // 9600   (divisible by 128)

    // 1) value = encoder @ W_val + b_val          (272000 x 256 x 256)
    gemm_wmma_f32<<<dim3(MV / BM, 256 / BN), 256, 0, stream>>>(
        encoder, W_val, b_val, value, MV, 256);

    // 2) off = hidden @ W_off + b_off             (9600 x 256 x 256)
    gemm_wmma_f32<<<dim3(MQ / BM, 256 / BN), 256, 0, stream>>>(
        hidden, W_off, b_off, offb, MQ, 256);

    // 3) aw logits = hidden @ W_attn + b_attn     (9600 x 256 x 128)
    gemm_wmma_f32<<<dim3(MQ / BM, 128 / BN), 256, 0, stream>>>(
        hidden, W_attn, b_attn, awb, MQ, 128);

    // 4) softmax + bilinear sampling + weighted sum -> mixed (B,Q,256)
    deform_sample<<<dim3((B_ * Q_ * H_) / 8), 256, 0, stream>>>(
        value, offb, awb, refpts, mixed);

    // 5) out = mixed @ W_out + b_out              (9600 x 256 x 256)
    gemm_wmma_f32<<<dim3(MQ / BM, 256 / BN), 256, 0, stream>>>(
        mixed, W_out, b_out, out, MQ, 256);
}